// MetricModel_43190191129169
// MI455X (gfx1250) — compile-verified
//
#include <hip/hip_runtime.h>

// Problem constants (from reference): B=4, D=128, H=W=128, C=16
#define B_   4
#define D_   128
#define HW_  16384      // H*W pixels per batch
#define C_   16

typedef float v2f __attribute__((ext_vector_type(2)));
typedef float v8f __attribute__((ext_vector_type(8)));

// Workspace layout (floats):
//  [0,     8192)  sums[b][c][d]   (un-normalized prototypes)
//  [8192,  8256)  counts[b][c]
//  [8256,  8264)  accum[b][2]     ({masked-CE sum, mask sum} per batch)
#define WS_SUMS   0
#define WS_COUNTS 8192
#define WS_ACCUM  8256
#define WS_TOTAL  8264

__global__ void k_zero(float* __restrict__ ws) {
    int i = blockIdx.x * 256 + threadIdx.x;
    if (i < WS_TOTAL) ws[i] = 0.0f;
}

// ---------------------------------------------------------------------------
// Prototype accumulation: protos^T(D x C) = X(D x P) @ onehot(P x C)
// via V_WMMA_F32_16X16X4_F32.  Grid: B*8 blocks of 256 threads (8 waves).
// Each wave owns 256 pixels -> 64 K-steps of 4 pixels, 8 M-tiles (D=128).
// Also computes per-class pixel counts with an LDS histogram.
// ---------------------------------------------------------------------------
__global__ void __launch_bounds__(256)
k_protos(const float* __restrict__ x, const int* __restrict__ label,
         float* __restrict__ ws) {
    float* sums   = ws + WS_SUMS;
    float* counts = ws + WS_COUNTS;

    const int b       = blockIdx.x >> 3;          // 8 blocks per batch
    const int blkInB  = blockIdx.x & 7;
    const int blockP0 = blkInB * 2048;            // 2048 pixels per block

    const int* __restrict__ lb = label + b * HW_;

    // ---- class histogram (counts[b][c]) ----
    __shared__ int hist[C_];
    if (threadIdx.x < C_) hist[threadIdx.x] = 0;
    __syncthreads();
#pragma unroll
    for (int i = 0; i < 8; ++i) {
        int p = blockP0 + threadIdx.x + i * 256;
        atomicAdd(&hist[(lb[p] - 1) & 15], 1);
    }
    __syncthreads();
    if (threadIdx.x < C_)
        atomicAdd(&counts[b * C_ + threadIdx.x], (float)hist[threadIdx.x]);

    // ---- WMMA one-hot matmul ----
    const int lane    = threadIdx.x & 31;
    const int waveId  = threadIdx.x >> 5;
    const int halfSel = lane >> 4;                // 0: lanes 0-15, 1: 16-31
    const int mrow    = lane & 15;                // A: M row (d); B: N col (class)
    const int koff    = halfSel * 2;              // this half's K offset
    const int waveP0  = blockP0 + waveId * 256;

    const float* __restrict__ xb = x + (size_t)b * D_ * HW_;

    v8f acc[8] = {};                              // 8 tiles of 16(d) x 16(c)

    for (int s = 0; s < 64; ++s) {
        const int p = waveP0 + s * 4;

        // B fragment: one-hot, rows K=koff..koff+1 held by this half-wave.
        int2 lv = *(const int2*)(lb + p + koff);
        v2f bm;
        bm[0] = (lv.x - 1 == mrow) ? 1.0f : 0.0f;
        bm[1] = (lv.y - 1 == mrow) ? 1.0f : 0.0f;

#pragma unroll
        for (int t = 0; t < 8; ++t) {
            // A fragment: x rows d = t*16 + mrow, K along contiguous pixels.
            const float* ap = xb + (size_t)(t * 16 + mrow) * HW_ + p + koff;
            v2f a = *(const v2f*)ap;
            acc[t] = __builtin_amdgcn_wmma_f32_16x16x4_f32(
                false, a, false, bm, (short)0, acc[t], false, false);
        }
    }

    // Reduce partial tiles into global sums[b][c][d].
#pragma unroll
    for (int t = 0; t < 8; ++t) {
#pragma unroll
        for (int r = 0; r < 8; ++r) {
            int d = t * 16 + r + halfSel * 8;     // C/D layout: VGPR r holds M=r / M=r+8
            atomicAdd(&sums[(b * C_ + mrow) * D_ + d], acc[t][r]);
        }
    }
}

// ---------------------------------------------------------------------------
// Fused L1-distance + min-shift + log-softmax CE + masked reduction.
// Grid: B*64 blocks of 256 threads; one thread per pixel.
// ---------------------------------------------------------------------------
__global__ void __launch_bounds__(256)
k_loss(const float* __restrict__ x, const int* __restrict__ label,
       const int* __restrict__ mask, float* __restrict__ ws) {
    __shared__ float smem[C_ * D_];               // 8 KB: protos, then reduction

    const int b = blockIdx.x >> 6;
    const int p = (blockIdx.x & 63) * 256 + threadIdx.x;

    const float* __restrict__ sums   = ws + WS_SUMS   + b * C_ * D_;
    const float* __restrict__ counts = ws + WS_COUNTS + b * C_;

    // Normalize prototypes into LDS: smem[c*128 + d]
#pragma unroll
    for (int i = 0; i < 8; ++i) {
        int idx = threadIdx.x + i * 256;
        smem[idx] = sums[idx] / counts[idx >> 7];
    }
    __syncthreads();

    // L1 distance to all 16 prototypes; x loads coalesced along p.
    const float* __restrict__ xb = x + (size_t)b * D_ * HW_ + p;
    float dist[C_];
#pragma unroll
    for (int c = 0; c < C_; ++c) dist[c] = 0.0f;

    for (int d = 0; d < D_; ++d) {
        float xv = xb[(size_t)d * HW_];
#pragma unroll
        for (int c = 0; c < C_; ++c)
            dist[c] += fabsf(xv - smem[c * D_ + d]);
    }

    // pd0 = exp(-dist); pd = pd0 - min; CE = lse(pd) - pd[local]
    const int local = label[b * HW_ + p] - 1;
    float pd[C_];
    float minv = 3.402823466e38f;
#pragma unroll
    for (int c = 0; c < C_; ++c) {
        pd[c] = __expf(-dist[c]);
        minv  = fminf(minv, pd[c]);
    }
    float maxv = -3.402823466e38f;
#pragma unroll
    for (int c = 0; c < C_; ++c) {
        pd[c] -= minv;
        maxv   = fmaxf(maxv, pd[c]);
    }
    float sum = 0.0f, pdl = 0.0f;
#pragma unroll
    for (int c = 0; c < C_; ++c) {
        sum += __expf(pd[c] - maxv);
        if (c == local) pdl = pd[c];
    }
    const float lse = maxv + __logf(sum);
    const float m   = (float)mask[b * HW_ + p];
    const float ce  = (lse - pdl) * m;

    // Block reduction of (ce, m)
    __syncthreads();
    smem[threadIdx.x]       = ce;
    smem[256 + threadIdx.x] = m;
    __syncthreads();
    for (int off = 128; off > 0; off >>= 1) {
        if (threadIdx.x < off) {
            smem[threadIdx.x]       += smem[threadIdx.x + off];
            smem[256 + threadIdx.x] += smem[256 + threadIdx.x + off];
        }
        __syncthreads();
    }
    if (threadIdx.x == 0) {
        atomicAdd(&ws[WS_ACCUM + b * 2 + 0], smem[0]);
        atomicAdd(&ws[WS_ACCUM + b * 2 + 1], smem[256]);
    }
}

__global__ void k_final(const float* __restrict__ ws, float* __restrict__ out) {
    if (threadIdx.x == 0) {
        float t = 0.0f;
#pragma unroll
        for (int b = 0; b < B_; ++b)
            t += ws[WS_ACCUM + b * 2] / ws[WS_ACCUM + b * 2 + 1];
        out[0] = t;
    }
}

extern "C" void kernel_launch(void* const* d_in, const int* in_sizes, int n_in,
                              void* d_out, int out_size, void* d_ws, size_t ws_size,
                              hipStream_t stream) {
    const float* x     = (const float*)d_in[0];   // [B,D,H,W] f32
    const int*   label = (const int*)d_in[1];     // [B,H,W]   i32 in [1,16]
    const int*   mask  = (const int*)d_in[2];     // [B,1,H,W] i32 in {0,1}
    float*       out   = (float*)d_out;           // scalar f32
    float*       ws    = (float*)d_ws;

    (void)in_sizes; (void)n_in; (void)out_size; (void)ws_size;

    k_zero  <<<(WS_TOTAL + 255) / 256, 256, 0, stream>>>(ws);
    k_protos<<<B_ * 8,  256, 0, stream>>>(x, label, ws);
    k_loss  <<<B_ * 64, 256, 0, stream>>>(x, label, mask, ws);
    k_final <<<1, 32, 0, stream>>>(ws, out);
}